// GCNAnomalyDetector_5866925326770
// MI455X (gfx1250) — compile-verified
//
#include <hip/hip_runtime.h>
#include <hip/hip_bf16.h>

typedef float v2f __attribute__((ext_vector_type(2)));
typedef float v8f __attribute__((ext_vector_type(8)));

#define TPB 256
#define WAVES_PER_BLOCK (TPB / 32)

// ---------------------------------------------------------------------------
// 1) Zero accumulators, init deg with the self-loop contribution (deg = 1).
// ---------------------------------------------------------------------------
__global__ __launch_bounds__(TPB) void k_init(float* __restrict__ agg1,
                                              float* __restrict__ agg2,
                                              float* __restrict__ deg, int N) {
  int i = blockIdx.x * TPB + threadIdx.x;
  int n16 = N * 16;
  if (i < n16) agg1[i] = 0.0f;
  if (i < N * 2) agg2[i] = 0.0f;
  if (i < N) deg[i] = 1.0f;  // self-loop counted analytically
}

// ---------------------------------------------------------------------------
// 2) deg[dst] += 1 over the E explicit edges (float, exact for these counts).
// ---------------------------------------------------------------------------
__global__ __launch_bounds__(TPB) void k_deg(const int* __restrict__ dst,
                                             float* __restrict__ deg, int E) {
  int e = blockIdx.x * TPB + threadIdx.x;
  if (e < E) atomicAdd(&deg[dst[e]], 1.0f);
}

// ---------------------------------------------------------------------------
// 3) dinv = deg > 0 ? rsqrt(deg) : 0   (in place)
// ---------------------------------------------------------------------------
__global__ __launch_bounds__(TPB) void k_dinv(float* __restrict__ deg, int N) {
  int i = blockIdx.x * TPB + threadIdx.x;
  if (i < N) {
    float d = deg[i];
    deg[i] = (d > 0.0f) ? rsqrtf(d) : 0.0f;
  }
}

// ---------------------------------------------------------------------------
// 4) norm[e] = dinv[src[e]] * dinv[dst[e]]  (reused by both layers)
// ---------------------------------------------------------------------------
__global__ __launch_bounds__(TPB) void k_norm(const int* __restrict__ src,
                                              const int* __restrict__ dst,
                                              const float* __restrict__ dinv,
                                              float* __restrict__ norm, int E) {
  int e = blockIdx.x * TPB + threadIdx.x;
  if (e < E) norm[e] = dinv[src[e]] * dinv[dst[e]];
}

// ---------------------------------------------------------------------------
// 5) h1_pre = x @ W1   [N,128] x [128,16] via V_WMMA_F32_16X16X4_F32.
//    One wave computes one 16x16 output tile; K=128 -> 32 WMMA steps.
//    W1 (8 KB) staged in LDS.
//    A 16x4 f32 layout: lane m = lane&15; VGPR0 = K(2*hi), VGPR1 = K(2*hi+1)
//    B 4x16  f32 layout: lane n = lane&15; same K split across lane halves
//    C/D 16x16 f32:      lane n = lane&15; VGPR v holds row v + 8*hi
// ---------------------------------------------------------------------------
__global__ __launch_bounds__(TPB) void k_gemm1(const float* __restrict__ x,
                                               const float* __restrict__ W1,
                                               float* __restrict__ h1p, int N) {
  __shared__ float w[128 * 16];
  for (int i = threadIdx.x; i < 128 * 16; i += TPB) w[i] = W1[i];
  __syncthreads();

  int wave = blockIdx.x * WAVES_PER_BLOCK + (threadIdx.x >> 5);
  int lane = threadIdx.x & 31;
  int m = lane & 15;
  int hi = lane >> 4;

  int tile = wave;
  int ntiles = (N + 15) >> 4;
  if (tile >= ntiles) return;  // whole-wave uniform exit

  int row = tile * 16 + m;
  int rowc = (row < N) ? row : (N - 1);  // clamp loads; EXEC stays all-1s
  const float* xr = x + (size_t)rowc * 128;

  v8f c = {};
#pragma unroll
  for (int k0 = 0; k0 < 128; k0 += 4) {
    int ka = k0 + 2 * hi;
    v2f a;
    a.x = xr[ka];
    a.y = xr[ka + 1];
    v2f b;
    b.x = w[ka * 16 + m];
    b.y = w[(ka + 1) * 16 + m];
    c = __builtin_amdgcn_wmma_f32_16x16x4_f32(false, a, false, b, (short)0, c,
                                              false, false);
  }

  // store: lane holds column n=m of rows (v + 8*hi)
#pragma unroll
  for (int v = 0; v < 8; ++v) {
    int r = tile * 16 + v + 8 * hi;
    if (r < N) h1p[(size_t)r * 16 + m] = c[v];
  }
}

// ---------------------------------------------------------------------------
// 6) Layer-1 edge aggregation: 16 lanes per edge, lane = feature.
//    Coalesced 64B gather of h1_pre[src], coalesced atomic adds to agg1[dst].
// ---------------------------------------------------------------------------
__global__ __launch_bounds__(TPB) void k_edge1(const int* __restrict__ src,
                                               const int* __restrict__ dst,
                                               const float* __restrict__ norm,
                                               const float* __restrict__ h1p,
                                               float* __restrict__ agg1, int E) {
  int t = blockIdx.x * TPB + threadIdx.x;
  int e = t >> 4;
  int f = t & 15;
  if (e >= E) return;
  int s = src[e];
  int d = dst[e];
  float v = h1p[(size_t)s * 16 + f] * norm[e];
  atomicAdd(&agg1[(size_t)d * 16 + f], v);
}

// ---------------------------------------------------------------------------
// 7) Node epilogue 1: add analytic self-loop (dinv^2 * h1_pre) + bias, ReLU.
//    agg1 becomes h1 in place (elementwise, safe).
// ---------------------------------------------------------------------------
__global__ __launch_bounds__(TPB) void k_node1(float* __restrict__ agg1,
                                               const float* __restrict__ h1p,
                                               const float* __restrict__ dinv,
                                               const float* __restrict__ b1,
                                               int N) {
  int idx = blockIdx.x * TPB + threadIdx.x;
  if (idx >= N * 16) return;
  int i = idx >> 4;
  int f = idx & 15;
  float dd = dinv[i] * dinv[i];
  float v = agg1[idx] + dd * h1p[idx] + b1[f];
  agg1[idx] = fmaxf(v, 0.0f);
}

// ---------------------------------------------------------------------------
// 8) h2_pre = h1 @ W2   [N,16] x [16,2], W2 zero-padded to 16x16 in LDS.
//    Same WMMA tiling; only output columns 0..1 are stored.
// ---------------------------------------------------------------------------
__global__ __launch_bounds__(TPB) void k_gemm2(const float* __restrict__ h1,
                                               const float* __restrict__ W2,
                                               float* __restrict__ h2p, int N) {
  __shared__ float w[16 * 16];
  for (int i = threadIdx.x; i < 16 * 16; i += TPB) {
    int k = i >> 4, n = i & 15;
    w[i] = (n < 2) ? W2[k * 2 + n] : 0.0f;
  }
  __syncthreads();

  int wave = blockIdx.x * WAVES_PER_BLOCK + (threadIdx.x >> 5);
  int lane = threadIdx.x & 31;
  int m = lane & 15;
  int hi = lane >> 4;

  int tile = wave;
  int ntiles = (N + 15) >> 4;
  if (tile >= ntiles) return;

  int row = tile * 16 + m;
  int rowc = (row < N) ? row : (N - 1);
  const float* hr = h1 + (size_t)rowc * 16;

  v8f c = {};
#pragma unroll
  for (int k0 = 0; k0 < 16; k0 += 4) {
    int ka = k0 + 2 * hi;
    v2f a;
    a.x = hr[ka];
    a.y = hr[ka + 1];
    v2f b;
    b.x = w[ka * 16 + m];
    b.y = w[(ka + 1) * 16 + m];
    c = __builtin_amdgcn_wmma_f32_16x16x4_f32(false, a, false, b, (short)0, c,
                                              false, false);
  }

  if (m < 2) {
#pragma unroll
    for (int v = 0; v < 8; ++v) {
      int r = tile * 16 + v + 8 * hi;
      if (r < N) h2p[(size_t)r * 2 + m] = c[v];
    }
  }
}

// ---------------------------------------------------------------------------
// 9) Layer-2 edge aggregation: 2 lanes per edge.
// ---------------------------------------------------------------------------
__global__ __launch_bounds__(TPB) void k_edge2(const int* __restrict__ src,
                                               const int* __restrict__ dst,
                                               const float* __restrict__ norm,
                                               const float* __restrict__ h2p,
                                               float* __restrict__ agg2, int E) {
  int t = blockIdx.x * TPB + threadIdx.x;
  int e = t >> 1;
  int f = t & 1;
  if (e >= E) return;
  int s = src[e];
  int d = dst[e];
  float v = h2p[(size_t)s * 2 + f] * norm[e];
  atomicAdd(&agg2[(size_t)d * 2 + f], v);
}

// ---------------------------------------------------------------------------
// 10) Final: self-loop + bias + log_softmax over the 2 classes.
// ---------------------------------------------------------------------------
__global__ __launch_bounds__(TPB) void k_final(const float* __restrict__ agg2,
                                               const float* __restrict__ h2p,
                                               const float* __restrict__ dinv,
                                               const float* __restrict__ b2,
                                               float* __restrict__ out, int N) {
  int i = blockIdx.x * TPB + threadIdx.x;
  if (i >= N) return;
  float dd = dinv[i] * dinv[i];
  float z0 = agg2[2 * i + 0] + dd * h2p[2 * i + 0] + b2[0];
  float z1 = agg2[2 * i + 1] + dd * h2p[2 * i + 1] + b2[1];
  float mx = fmaxf(z0, z1);
  float lse = mx + logf(expf(z0 - mx) + expf(z1 - mx));
  out[2 * i + 0] = z0 - lse;
  out[2 * i + 1] = z1 - lse;
}

// ---------------------------------------------------------------------------
// launch
// ---------------------------------------------------------------------------
extern "C" void kernel_launch(void* const* d_in, const int* in_sizes, int n_in,
                              void* d_out, int out_size, void* d_ws,
                              size_t ws_size, hipStream_t stream) {
  const float* x = (const float*)d_in[0];
  const int* eidx = (const int*)d_in[1];
  const float* W1 = (const float*)d_in[2];
  const float* b1 = (const float*)d_in[3];
  const float* W2 = (const float*)d_in[4];
  const float* b2 = (const float*)d_in[5];
  float* out = (float*)d_out;

  const int N = in_sizes[0] / 128;
  const int E = in_sizes[1] / 2;
  const int* src = eidx;       // edge_index[0]
  const int* dst = eidx + E;   // edge_index[1]

  // workspace layout (floats): deg/dinv[N] | norm[E] | h1p[16N] | agg1[16N] | h2p[2N] | agg2[2N]
  float* ws = (float*)d_ws;
  float* deg = ws;
  float* norm = deg + N;
  float* h1p = norm + E;
  float* agg1 = h1p + (size_t)16 * N;
  float* h2p = agg1 + (size_t)16 * N;
  float* agg2 = h2p + (size_t)2 * N;

  const int ntiles = (N + 15) / 16;
  const int gemm_blocks = (ntiles + WAVES_PER_BLOCK - 1) / WAVES_PER_BLOCK;

  k_init<<<(N * 16 + TPB - 1) / TPB, TPB, 0, stream>>>(agg1, agg2, deg, N);
  k_deg<<<(E + TPB - 1) / TPB, TPB, 0, stream>>>(dst, deg, E);
  k_dinv<<<(N + TPB - 1) / TPB, TPB, 0, stream>>>(deg, N);
  k_norm<<<(E + TPB - 1) / TPB, TPB, 0, stream>>>(src, dst, deg, norm, E);

  k_gemm1<<<gemm_blocks, TPB, 0, stream>>>(x, W1, h1p, N);
  k_edge1<<<((long long)E * 16 + TPB - 1) / TPB, TPB, 0, stream>>>(src, dst, norm, h1p, agg1, E);
  k_node1<<<(N * 16 + TPB - 1) / TPB, TPB, 0, stream>>>(agg1, h1p, deg, b1, N);

  k_gemm2<<<gemm_blocks, TPB, 0, stream>>>(agg1, W2, h2p, N);
  k_edge2<<<((long long)E * 2 + TPB - 1) / TPB, TPB, 0, stream>>>(src, dst, norm, h2p, agg2, E);
  k_final<<<(N + TPB - 1) / TPB, TPB, 0, stream>>>(agg2, h2p, deg, b2, out, N);
}